// GCN_54228257079640
// MI455X (gfx1250) — compile-verified
//
#include <hip/hip_runtime.h>
#include <hip/hip_bf16.h>

// ---------------------------------------------------------------------------
// GCN forward for MI455X (gfx1250).
// Bottleneck is the edge gather/scatter (~10 GB, L2-resident: 154 MB working
// set < 192 MB L2). GEMMs use native f32 WMMA (V_WMMA_F32_16X16X4_F32) so no
// precision is sacrificed — they are far from the roofline anyway.
// ---------------------------------------------------------------------------

#define N_NODES  100000
#define N_EDGES  3200000
#define N_GRAPHS 128
#define DH       128
#define D_OUT    10

typedef __attribute__((ext_vector_type(2))) float v2f;
typedef __attribute__((ext_vector_type(8))) float v8f;

// ---------------------------------------------------------------- fill ------
__global__ void fill_f32(float* __restrict__ p, float v, size_t n) {
    size_t i = (size_t)blockIdx.x * blockDim.x + threadIdx.x;
    size_t stride = (size_t)gridDim.x * blockDim.x;
    for (; i < n; i += stride) p[i] = v;
}

// ------------------------------------------------------------- degree -------
__global__ void deg_kernel(const int* __restrict__ dst, float* __restrict__ deg, int nE) {
    int e = blockIdx.x * blockDim.x + threadIdx.x;
    if (e < nE) atomicAdd(&deg[dst[e]], 1.0f);
}

// deg -> (dinv_sqrt in place, inv_deg)
__global__ void norm_kernel(float* __restrict__ dinv, float* __restrict__ invd, int n) {
    int i = blockIdx.x * blockDim.x + threadIdx.x;
    if (i < n) {
        float g = dinv[i];          // currently holds degree (>= 1)
        invd[i] = 1.0f / g;
        dinv[i] = rsqrtf(g);
    }
}

// ------------------------------------------------ WMMA f32 GEMM (K=N=128) ---
// C[M,128] = act( A[M,128] @ W[128,128] + bias )
// One block = 256 threads = 8 wave32; block handles one 16-row M tile,
// wave w handles N-tile [16w, 16w+16). A tile staged in LDS (padded stride
// 132 floats -> conflict-free 16-lane reads). M must be a multiple of 16.
__launch_bounds__(256)
__global__ void gemm_wmma_f32(const float* __restrict__ A,
                              const float* __restrict__ W,
                              const float* __restrict__ bias,   // may be null
                              float* __restrict__ C,
                              int do_relu) {
    __shared__ float As[16 * 132];

    const int m0 = blockIdx.x * 16;

    // cooperative load of 16x128 A tile (512 float4 / 256 threads)
    const float4* asrc = (const float4*)(A + (size_t)m0 * DH);
    for (int f = threadIdx.x; f < 512; f += 256) {
        int row = f >> 5;          // 0..15
        int c4  = f & 31;          // 0..31
        *(float4*)&As[row * 132 + c4 * 4] = asrc[f];
    }
    __syncthreads();

    const int wave = threadIdx.x >> 5;
    const int lane = threadIdx.x & 31;
    const int n0   = wave * 16;
    const int half = lane >> 4;    // 0: lanes 0-15, 1: lanes 16-31
    const int idx  = lane & 15;

    v8f c = {};
    // K = 128 in steps of 4: 32 chained v_wmma_f32_16x16x4_f32
    #pragma unroll
    for (int k0 = 0; k0 < DH; k0 += 4) {
        const int ka = k0 + 2 * half;
        v2f a, b;
        // A 16x4 fragment: lanes 0-15 -> rows 0-15, VGPR pair = (K, K+1),
        // upper half-lanes carry K+2, K+3.
        a.x = As[idx * 132 + ka];
        a.y = As[idx * 132 + ka + 1];
        // B 4x16 fragment (row striped across lanes, K split at lane 16)
        b.x = W[(size_t)ka * DH + n0 + idx];
        b.y = W[(size_t)(ka + 1) * DH + n0 + idx];
        c = __builtin_amdgcn_wmma_f32_16x16x4_f32(
                /*neg_a=*/false, a, /*neg_b=*/false, b,
                /*c_mod=*/(short)0, c, /*reuse_a=*/false, /*reuse_b=*/false);
    }

    // C/D layout: VGPR j -> row j (lanes 0-15) / row j+8 (lanes 16-31)
    #pragma unroll
    for (int j = 0; j < 8; ++j) {
        int row = m0 + j + 8 * half;
        int col = n0 + idx;
        float v = c[j];
        if (bias)    v += bias[col];
        if (do_relu) v = fmaxf(v, 0.0f);
        C[(size_t)row * DH + col] = v;
    }
}

// -------------------------------------------------- edge gather/scatter -----
// One wave32 per edge; lane handles 4 consecutive dims (float4 gather,
// 4x global_atomic_add_f32 scatter). xw (51 MB) is L2-resident.
__launch_bounds__(256)
__global__ void edge_scatter(const float* __restrict__ xw,
                             const int* __restrict__ src,
                             const int* __restrict__ dst,
                             const float* __restrict__ dinv,
                             float* __restrict__ agg, int nE) {
    int t = blockIdx.x * blockDim.x + threadIdx.x;
    int e = t >> 5;
    if (e >= nE) return;
    int lane = t & 31;
    int s = src[e];
    int d = dst[e];
    float w = dinv[s] * dinv[d];           // norm_e
    float4 v = ((const float4*)(xw + (size_t)s * DH))[lane];
    float* ag = agg + (size_t)d * DH + lane * 4;
    atomicAdd(ag + 0, v.x * w);
    atomicAdd(ag + 1, v.y * w);
    atomicAdd(ag + 2, v.z * w);
    atomicAdd(ag + 3, v.w * w);
}

// --------------------------------------------- epilogue: self-loop + pool ---
// One wave32 per node: h = relu(agg + xw*inv_deg + b); pooled[batch[n]] += h
__launch_bounds__(256)
__global__ void node_post(const float* __restrict__ xw,
                          const float* __restrict__ agg,
                          const float* __restrict__ invd,
                          const float* __restrict__ bias,
                          const int* __restrict__ batch,
                          float* __restrict__ h,
                          float* __restrict__ pooled, int nN) {
    int t = blockIdx.x * blockDim.x + threadIdx.x;
    int node = t >> 5;
    if (node >= nN) return;
    int lane = t & 31;
    float id = invd[node];
    int g = batch[node];
    float4 a  = ((const float4*)(agg + (size_t)node * DH))[lane];
    float4 xv = ((const float4*)(xw  + (size_t)node * DH))[lane];
    float4 bv = ((const float4*)bias)[lane];
    float4 hv;
    hv.x = fmaxf(a.x + xv.x * id + bv.x, 0.0f);
    hv.y = fmaxf(a.y + xv.y * id + bv.y, 0.0f);
    hv.z = fmaxf(a.z + xv.z * id + bv.z, 0.0f);
    hv.w = fmaxf(a.w + xv.w * id + bv.w, 0.0f);
    ((float4*)(h + (size_t)node * DH))[lane] = hv;
    float* p = pooled + (size_t)g * DH + lane * 4;
    atomicAdd(p + 0, hv.x);
    atomicAdd(p + 1, hv.y);
    atomicAdd(p + 2, hv.z);
    atomicAdd(p + 3, hv.w);
}

// ------------------------------------------------------------- lin2 head ----
// out[128,10] = pfc[128,128] @ lin2_w[128,10] + lin2_b  (tiny: plain VALU)
__global__ void lin2_kernel(const float* __restrict__ pfc,
                            const float* __restrict__ w2,
                            const float* __restrict__ b2,
                            float* __restrict__ out) {
    int t = blockIdx.x * blockDim.x + threadIdx.x;
    if (t >= N_GRAPHS * D_OUT) return;
    int r = t / D_OUT, c = t % D_OUT;
    float acc = b2[c];
    #pragma unroll 4
    for (int k = 0; k < DH; ++k)
        acc = fmaf(pfc[r * DH + k], w2[k * D_OUT + c], acc);
    out[t] = acc;
}

// ---------------------------------------------------------------------------
extern "C" void kernel_launch(void* const* d_in, const int* in_sizes, int n_in,
                              void* d_out, int out_size, void* d_ws, size_t ws_size,
                              hipStream_t stream) {
    (void)in_sizes; (void)n_in; (void)out_size; (void)ws_size;

    const float* x       = (const float*)d_in[0];
    const int*   src     = (const int*)d_in[1];              // edge_index[0]
    const int*   dst     = ((const int*)d_in[1]) + N_EDGES;  // edge_index[1]
    const int*   batch   = (const int*)d_in[2];
    const float* W[3]    = { (const float*)d_in[3], (const float*)d_in[5], (const float*)d_in[7] };
    const float* B[3]    = { (const float*)d_in[4], (const float*)d_in[6], (const float*)d_in[8] };
    const float* lin1_w  = (const float*)d_in[9];
    const float* lin1_b  = (const float*)d_in[10];
    const float* lin2_w  = (const float*)d_in[11];
    const float* lin2_b  = (const float*)d_in[12];

    float* out = (float*)d_out;                       // [128*10]
    float* pooled[3] = {                              // [128*128] each
        out + N_GRAPHS * D_OUT,
        out + N_GRAPHS * D_OUT + N_GRAPHS * DH,
        out + N_GRAPHS * D_OUT + 2 * N_GRAPHS * DH
    };

    // workspace layout
    const size_t NODE_BUF = (size_t)N_NODES * DH;     // floats
    float* bufA = (float*)d_ws;                       // xw
    float* bufB = bufA + NODE_BUF;                    // agg
    float* bufC = bufB + NODE_BUF;                    // h
    float* dinv = bufC + NODE_BUF;                    // deg -> dinv_sqrt
    float* invd = dinv + N_NODES;
    float* pfc  = invd + N_NODES;                     // [128,128] lin1 output

    const int TB = 256;

    // zero outputs (d_out is poisoned by harness; pooled built via atomics)
    {
        size_t n = (size_t)N_GRAPHS * D_OUT + 3 * (size_t)N_GRAPHS * DH;
        fill_f32<<<1024, TB, 0, stream>>>(out, 0.0f, n);
    }

    // degree (self-loop => start at 1), then dinv_sqrt / inv_deg
    fill_f32<<<1024, TB, 0, stream>>>(dinv, 1.0f, (size_t)N_NODES);
    deg_kernel<<<(N_EDGES + TB - 1) / TB, TB, 0, stream>>>(dst, dinv, N_EDGES);
    norm_kernel<<<(N_NODES + TB - 1) / TB, TB, 0, stream>>>(dinv, invd, N_NODES);

    const int gemm_blocks = N_NODES / 16;             // 6250, exact
    const int edge_blocks = N_EDGES * 32 / TB;        // 400000, exact
    const int node_blocks = N_NODES * 32 / TB;        // 12500, exact

    const float* h_in = x;
    for (int layer = 0; layer < 3; ++layer) {
        // xw = h_in @ W[layer]
        gemm_wmma_f32<<<gemm_blocks, TB, 0, stream>>>(h_in, W[layer], nullptr, bufA, 0);
        // agg = 0
        fill_f32<<<2048, TB, 0, stream>>>(bufB, 0.0f, NODE_BUF);
        // agg += norm_e * xw[src]  (scatter to dst)
        edge_scatter<<<edge_blocks, TB, 0, stream>>>(bufA, src, dst, dinv, bufB, N_EDGES);
        // h = relu(agg + xw*inv_deg + b); pooled[layer] += h per graph
        node_post<<<node_blocks, TB, 0, stream>>>(bufA, bufB, invd, B[layer], batch,
                                                  bufC, pooled[layer], N_NODES);
        h_in = bufC;
    }

    // pfc = relu(pooled3 @ lin1_w + lin1_b)   (M=128 -> 8 WMMA blocks)
    gemm_wmma_f32<<<N_GRAPHS / 16, TB, 0, stream>>>(pooled[2], lin1_w, lin1_b, pfc, 1);

    // out = pfc @ lin2_w + lin2_b
    lin2_kernel<<<(N_GRAPHS * D_OUT + TB - 1) / TB, TB, 0, stream>>>(pfc, lin2_w, lin2_b, out);
}